// LinOSS_6347961663978
// MI455X (gfx1250) — compile-verified
//
#include <hip/hip_runtime.h>
#include <math.h>

#define NH 8
#define HD 64
#define TT 4096
#define BB 4
#define DT 0.05f
#define CH 16                    // timesteps per chunk ( == WMMA M,N )
#define NCHUNK (TT / CH)
#define BPP 4                    // blocks per (b,h) pair (64 cols split 4 ways)
#define NTHR 256                 // 8 wave32
#define SKV 68                   // padded step stride (floats) for staged k/v/q
#define STG_V (16 * SKV)         // 1088
#define STG_Q (2 * 16 * SKV)     // 2176
#define STG_B (3 * 16 * SKV)     // 3264
#define BUFF  (3 * 16 * SKV + 16)// 3280 floats per staging buffer
#define YSTEP 68                 // padded y-snapshot step stride
#define YCOL  (16 * YSTEP + 4)   // 1092: column stride (pad kills 2-way conflicts)
#define YSZ   (16 * YCOL)        // 17472 floats (~68 KB)

#if defined(__gfx1250__) && __has_builtin(__builtin_amdgcn_wmma_f32_16x16x4_f32)
#define USE_WMMA 1
#else
#define USE_WMMA 0
#endif

typedef __attribute__((ext_vector_type(2))) float v2f;
typedef __attribute__((ext_vector_type(8))) float v8f;

// ---- CDNA5 async global->LDS copies (ASYNCcnt) ----
__device__ __forceinline__ void async_b128(unsigned lds_off, unsigned long long src) {
    asm volatile("global_load_async_to_lds_b128 %0, %1, off"
                 :: "v"(lds_off), "v"(src) : "memory");
}
__device__ __forceinline__ void async_b32(unsigned lds_off, unsigned long long src) {
    asm volatile("global_load_async_to_lds_b32 %0, %1, off"
                 :: "v"(lds_off), "v"(src) : "memory");
}

// ---- wave32 XOR butterfly via ds_swizzle (group-of-32: and=0x1f, or=0, xor=M) ----
template <int M>
__device__ __forceinline__ float swz_xor(float v) {
    return __int_as_float(
        __builtin_amdgcn_ds_swizzle(__float_as_int(v), (M << 10) | 0x1f));
}

__global__ __launch_bounds__(NTHR) void linoss_scan(
    const float* __restrict__ q, const float* __restrict__ k,
    const float* __restrict__ v, const float* __restrict__ beta,
    const float* __restrict__ y0p, const float* __restrict__ z0p,
    const float* __restrict__ ow, const float* __restrict__ od,
    float* __restrict__ out)
{
    __shared__ __align__(16) float sstage[2 * BUFF];
#if USE_WMMA
    __shared__ __align__(16) float ysnap[YSZ];
#endif

    const int tid  = threadIdx.x;
    const int pair = blockIdx.x >> 2;          // (b,h)
    const int b    = pair >> 3;
    const int h    = pair & 7;
    const int eoff = (blockIdx.x & 3) << 4;    // column offset of this block
    const int el   = tid >> 4;                 // block-local column [0,16)
    const int r    = tid & 15;                 // 16 lanes per column
    const int eg   = eoff + el;                // global column e
    const int d0   = r << 2;                   // 4 owned rows

    const float* kb = k    + (size_t)b * TT * NH * HD + (size_t)h * HD;
    const float* vb = v    + (size_t)b * TT * NH * HD + (size_t)h * HD;
    const float* qb = q    + (size_t)b * TT * NH * HD + (size_t)h * HD;
    const float* bb = beta + (size_t)b * TT * NH + h;
    float*       obp = out + (size_t)b * TT * NH * HD + (size_t)h * HD;
#if !USE_WMMA
    float*       ob  = obp + eg;
#endif

    // dtA = dt*exp(2*(c - softplus(c - w)));  dtB = dt*sigmoid(damp)/dt = sigmoid(damp)
    float y[4], z[4], dtA[4], dtB[4];
    const float lwm = logf(2.0f / DT * 0.99f);
#pragma unroll
    for (int i = 0; i < 4; ++i) {
        const int idx = (h * HD + d0 + i) * HD + eg;
        y[i] = y0p[idx];
        z[i] = z0p[idx];
        float x  = lwm - ow[idx];
        float sp = (x > 30.0f) ? x : log1pf(expf(x));
        dtA[i] = DT * expf(2.0f * (lwm - sp));
        dtB[i] = 1.0f / (1.0f + expf(-od[idx]));
    }

    const unsigned stg0 = (unsigned)(size_t)(void*)sstage; // low 32 bits = LDS offset
    const int s16 = tid >> 4, g = tid & 15;
    const int wv  = tid >> 5, ln = tid & 31;

    auto prefetch = [&](int c, int buf) {
        const int    t0   = c * CH;
        const size_t srco = (size_t)(t0 + s16) * (NH * HD) + 4 * g;
        const unsigned dk = stg0 + (unsigned)(buf * BUFF * 4 + s16 * (SKV * 4) + g * 16);
        async_b128(dk,             (unsigned long long)(size_t)(kb + srco));
        async_b128(dk + STG_V * 4, (unsigned long long)(size_t)(vb + srco));
        async_b128(dk + STG_Q * 4, (unsigned long long)(size_t)(qb + srco));
        if (ln < 2) {  // 2 betas/wave -> uniform 4 async ops per wave per chunk
            const int ti = t0 + wv * 2 + ln;
            async_b32(stg0 + (unsigned)(buf * BUFF * 4 + STG_B * 4 + (wv * 2 + ln) * 4),
                      (unsigned long long)(size_t)(bb + (size_t)ti * NH));
        }
    };

    prefetch(0, 0);

    for (int c = 0; c < NCHUNK; ++c) {
        const int buf = c & 1;
        if (c + 1 < NCHUNK) {
            prefetch(c + 1, buf ^ 1);
            asm volatile("s_wait_asynccnt 4" ::: "memory");
        } else {
            asm volatile("s_wait_asynccnt 0" ::: "memory");
        }
        __syncthreads();

        const float* kk = sstage + buf * BUFF;
        const float* vv = kk + STG_V;
        const float* qq = kk + STG_Q;
        const float* bt = kk + STG_B;

#pragma unroll
        for (int s = 0; s < CH; ++s) {
            // delta-rule readout: p = sum_d k[d]*y[d][e]  (16 lanes x 4 rows)
            const float4 kc = *(const float4*)(kk + s * SKV + d0);
            float p = kc.x * y[0] + kc.y * y[1] + kc.z * y[2] + kc.w * y[3];
            p += swz_xor<1>(p); p += swz_xor<2>(p); p += swz_xor<4>(p); p += swz_xor<8>(p);

            const float vt  = vv[s * SKV + eg] - p;
            const float c0s = (DT * bt[s]) * vt;            // dt*beta*vt
            const float kd[4] = {kc.x, kc.y, kc.z, kc.w};
#pragma unroll
            for (int i = 0; i < 4; ++i) {
                const float zo = z[i];
                float zn = fmaf(c0s, kd[i], zo);            // + dt*b*k[d]*vt[e]
                zn = fmaf(-dtA[i], y[i], zn);               // - dt*A o y
                zn = fmaf(-dtB[i], zo,   zn);               // - dt*B o z
                z[i] = zn;
                y[i] = fmaf(DT, zn, y[i]);                  // y += dt*z
            }

#if USE_WMMA
            // snapshot y_s for the batched WMMA readout (off the critical path)
            *(float4*)(ysnap + el * YCOL + s * YSTEP + d0) =
                make_float4(y[0], y[1], y[2], y[3]);
#else
            const float4 qc = *(const float4*)(qq + s * SKV + d0);
            float p2 = qc.x * y[0] + qc.y * y[1] + qc.z * y[2] + qc.w * y[3];
            p2 += swz_xor<1>(p2); p2 += swz_xor<2>(p2);
            p2 += swz_xor<4>(p2); p2 += swz_xor<8>(p2);
            if (r == 0) ob[(size_t)(c * CH + s) * (NH * HD)] = 0.125f * p2;
#endif
        }

#if USE_WMMA
        {   // Batched readout: per owned column e, G = Q(16x64) x Ysnap_e^T(64x16)
            // via 16 chained V_WMMA_F32_16X16X4_F32; out_s = 0.125 * G[s,s].
            // A frag (16x4): lane l -> A[l%16, k0+2*(l/16)+{0,1}]   (ds_load_b64)
            // B frag (4x16): lane l -> B[k0+2*(l/16)+{0,1}, l%16]   (ds_load_b64)
            const int lm = ln & 15, lh = ln >> 4;
            const float* qrow = qq + lm * SKV + 2 * lh;
            const float* yr0  = ysnap + (wv * 2) * YCOL + lm * YSTEP + 2 * lh;
            const float* yr1  = yr0 + YCOL;
            v8f acc0 = {0.f, 0.f, 0.f, 0.f, 0.f, 0.f, 0.f, 0.f};
            v8f acc1 = acc0;
#pragma unroll
            for (int ks = 0; ks < 16; ++ks) {
                const v2f a  = *(const v2f*)(qrow + 4 * ks);
                const v2f b0 = *(const v2f*)(yr0  + 4 * ks);
                const v2f b1 = *(const v2f*)(yr1  + 4 * ks);
                acc0 = __builtin_amdgcn_wmma_f32_16x16x4_f32(
                    false, a, false, b0, (short)0, acc0, false, false);
                acc1 = __builtin_amdgcn_wmma_f32_16x16x4_f32(
                    false, a, false, b1, (short)0, acc1, false, false);
            }
            // diagonal (s,s): VGPR s&7, lane s (s<8) or 16+s (s>=8)
            const int ii = ln & 7;
            float o0 = acc0[0], o1 = acc1[0];
#pragma unroll
            for (int j = 1; j < 8; ++j) {
                if (ii == j) { o0 = acc0[j]; o1 = acc1[j]; }
            }
            if (ln < 8 || ln >= 24) {
                const int s_ = ii + ((ln & 16) >> 1);       // +8 for upper half
                const size_t t = (size_t)(c * CH + s_);
                obp[t * (NH * HD) + eoff + wv * 2    ] = 0.125f * o0;
                obp[t * (NH * HD) + eoff + wv * 2 + 1] = 0.125f * o1;
            }
        }
#endif
        __syncthreads();   // all waves done with buf before it is overwritten
    }
}

extern "C" void kernel_launch(void* const* d_in, const int* in_sizes, int n_in,
                              void* d_out, int out_size, void* d_ws, size_t ws_size,
                              hipStream_t stream) {
    const float* q    = (const float*)d_in[0];
    const float* k    = (const float*)d_in[1];
    const float* v    = (const float*)d_in[2];
    const float* beta = (const float*)d_in[3];
    const float* y0p  = (const float*)d_in[4];
    const float* z0p  = (const float*)d_in[5];
    const float* ow   = (const float*)d_in[6];
    const float* od   = (const float*)d_in[7];
    float* out = (float*)d_out;

    dim3 grid(BB * NH * BPP);   // 128 blocks: 32 (b,h) pairs x 4 column splits
    dim3 block(NTHR);           // 8 wave32
    hipLaunchKernelGGL(linoss_scan, grid, block, 0, stream,
                       q, k, v, beta, y0p, z0p, ow, od, out);
}